// FC_45354854645899
// MI455X (gfx1250) — compile-verified
//
#include <hip/hip_runtime.h>
#include <math.h>

#define NUM_SEGS 64
#define D_IN     128
#define D_H      512
#define D_OUT    256

typedef __attribute__((ext_vector_type(2))) float v2f;
typedef __attribute__((ext_vector_type(4))) float v4f;
typedef __attribute__((ext_vector_type(8))) float v8f;

// ---------------------------------------------------------------------------
// Kernel 0: init pooled accumulator to -inf (d_ws is not re-poisoned between
// replays, so we must re-init every call for determinism).
// ---------------------------------------------------------------------------
__global__ void init_pooled_kernel(float* __restrict__ pooled) {
    int i = blockIdx.x * blockDim.x + threadIdx.x;
    if (i < NUM_SEGS * D_IN) pooled[i] = -INFINITY;
}

// ---------------------------------------------------------------------------
// Kernel 1: streaming segment-max (the only memory-heavy phase: ~1.05 GB).
// One wave per row: 32 lanes x float4 = 128 contiguous floats (512B coalesced,
// non-temporal: single-use stream, don't pollute the 192MB L2).
// Accumulate into a 64x128 f32 LDS tile (32 KB) with LDS float-max atomics,
// then flush once per block with global float-max atomics (8192 per block).
// float max is associative/commutative -> result is deterministic.
// ---------------------------------------------------------------------------
__global__ void __launch_bounds__(1024)
seg_max_kernel(const int* __restrict__ coords,
               const float* __restrict__ feats,
               float* __restrict__ pooled, int N) {
    __shared__ float smax[NUM_SEGS * D_IN];   // 32 KB

    for (int i = threadIdx.x; i < NUM_SEGS * D_IN; i += blockDim.x)
        smax[i] = -INFINITY;
    __syncthreads();

    const int lane  = threadIdx.x & 31;
    const int wave  = threadIdx.x >> 5;
    const int wpb   = blockDim.x >> 5;                      // waves per block (32)
    const long long stride = (long long)gridDim.x * wpb;    // total waves

    for (long long row = (long long)blockIdx.x * wpb + wave; row < N; row += stride) {
        const int seg = __builtin_nontemporal_load(coords + row * 4) & (NUM_SEGS - 1);
        const v4f  v  = __builtin_nontemporal_load((const v4f*)(feats + row * D_IN + lane * 4));
        float* dst = &smax[seg * D_IN + lane * 4];
        __hip_atomic_fetch_max(dst + 0, v.x, __ATOMIC_RELAXED, __HIP_MEMORY_SCOPE_WORKGROUP);
        __hip_atomic_fetch_max(dst + 1, v.y, __ATOMIC_RELAXED, __HIP_MEMORY_SCOPE_WORKGROUP);
        __hip_atomic_fetch_max(dst + 2, v.z, __ATOMIC_RELAXED, __HIP_MEMORY_SCOPE_WORKGROUP);
        __hip_atomic_fetch_max(dst + 3, v.w, __ATOMIC_RELAXED, __HIP_MEMORY_SCOPE_WORKGROUP);
    }
    __syncthreads();

    for (int i = threadIdx.x; i < NUM_SEGS * D_IN; i += blockDim.x) {
        float m = smax[i];
        if (m > -INFINITY)
            __hip_atomic_fetch_max(&pooled[i], m, __ATOMIC_RELAXED, __HIP_MEMORY_SCOPE_AGENT);
    }
}

// ---------------------------------------------------------------------------
// WMMA f32 GEMM:  C[M,N] = A[M,K] @ B[K,N] + bias[N]   (all row-major f32)
// One wave per 16x16 output tile, K-loop in steps of 4 using
// V_WMMA_F32_16X16X4_F32 (full f32 precision, matches reference).
//
// Fragment layouts (ISA 7.12.2, wave32):
//   A 16x4 : lanes 0-15  -> M=lane,    reg0=K0, reg1=K1
//            lanes 16-31 -> M=lane-16, reg0=K2, reg1=K3
//   B 4x16 : reg0: lanes 0-15 row K0, lanes 16-31 row K2
//            reg1: lanes 0-15 row K1, lanes 16-31 row K3   (N = lane&15)
//   C/D    : reg v: lanes 0-15 -> M=v, lanes 16-31 -> M=v+8 (N = lane&15)
// ---------------------------------------------------------------------------
template <int M, int N, int K>
__global__ void __launch_bounds__(256)
gemm_wmma_f32(const float* __restrict__ A, const float* __restrict__ B,
              const float* __restrict__ bias,
              float* __restrict__ C, float* __restrict__ C2) {
    const int lane   = threadIdx.x & 31;
    const int waveId = (blockIdx.x * blockDim.x + threadIdx.x) >> 5;
    const int tilesN = N / 16;
    const int m0 = (waveId / tilesN) * 16;
    const int n0 = (waveId % tilesN) * 16;

    const int l  = lane & 15;   // row (A) / col (B,C)
    const int hi = lane >> 4;   // half-wave select

    v8f c = {};
    for (int k0 = 0; k0 < K; k0 += 4) {
        const int ka = k0 + (hi << 1);
        v2f a, b;
        a.x = A[(m0 + l) * K + ka];
        a.y = A[(m0 + l) * K + ka + 1];
        b.x = B[(ka + 0) * N + n0 + l];
        b.y = B[(ka + 1) * N + n0 + l];
        c = __builtin_amdgcn_wmma_f32_16x16x4_f32(
                /*neg_a=*/false, a, /*neg_b=*/false, b,
                /*c_mod=*/(short)0, c, /*reuse_a=*/false, /*reuse_b=*/false);
    }

    const int col = n0 + l;
    const float bv = bias[col];
#pragma unroll
    for (int v = 0; v < 8; ++v) {
        const int row = m0 + v + hi * 8;
        const float out = c[v] + bv;
        C[row * N + col] = out;
        if (C2) C2[row * N + col] = out;
    }
}

// ---------------------------------------------------------------------------
// Launch: init -> segment-max -> GEMM1 (pooled@w1+b1) -> GEMM2 (h@w2+b2),
// GEMM2 writes emb into both halves of d_out (reference returns (emb, emb)).
// ---------------------------------------------------------------------------
extern "C" void kernel_launch(void* const* d_in, const int* in_sizes, int n_in,
                              void* d_out, int out_size, void* d_ws, size_t ws_size,
                              hipStream_t stream) {
    const int*   coords = (const int*)  d_in[0];
    const float* feats  = (const float*)d_in[1];
    const float* w1     = (const float*)d_in[2];
    const float* b1     = (const float*)d_in[3];
    const float* w2     = (const float*)d_in[4];
    const float* b2     = (const float*)d_in[5];
    float* out = (float*)d_out;

    const int N = in_sizes[0] / 4;

    float* pooled = (float*)d_ws;                 // 64 x 128  (32 KB)
    float* h      = pooled + NUM_SEGS * D_IN;     // 64 x 512  (128 KB)

    init_pooled_kernel<<<(NUM_SEGS * D_IN + 255) / 256, 256, 0, stream>>>(pooled);

    // 256 blocks x 32 waves = 8192 concurrent rows in flight; flush atomics
    // stay at 256 * 8192 = 2.1M (well under the 45us streaming time).
    seg_max_kernel<<<256, 1024, 0, stream>>>(coords, feats, pooled, N);

    // GEMM1: (64x128)@(128x512): 4*32 = 128 tiles -> 128 waves -> 16 blocks
    gemm_wmma_f32<NUM_SEGS, D_H, D_IN><<<16, 256, 0, stream>>>(pooled, w1, b1, h, nullptr);

    // GEMM2: (64x512)@(512x256): 4*16 = 64 tiles -> 64 waves -> 8 blocks
    gemm_wmma_f32<NUM_SEGS, D_OUT, D_H><<<8, 256, 0, stream>>>(h, w2, b2,
                                                               out, out + NUM_SEGS * D_OUT);
}